// TopKRouter_80444737454352
// MI455X (gfx1250) — compile-verified
//
#include <hip/hip_runtime.h>

// CDNA5 (gfx1250) MoE top-2 router: logits = x @ W^T via V_WMMA_F32_16X16X4_F32,
// async global->LDS double-buffered staging (ASYNCcnt), then exact softmax +
// top-2 + renorm. wave32, WGP mode.

typedef float v2f __attribute__((ext_vector_type(2)));
typedef float v8f __attribute__((ext_vector_type(8)));

#define ROUTER_D   2048   // d_model
#define ROUTER_E   64     // n_experts
#define TOK_PER_WG 64     // 4 waves x 16 tokens
#define BK         64     // K-block staged in LDS
#define LSTR       68     // padded LDS stride (dwords) -> conflict-free frag reads
#define NSTAGE     (ROUTER_D / BK)

// Async DMA of 16 bytes global -> LDS, tracked by ASYNCcnt (no VGPR round-trip).
__device__ __forceinline__ void async_copy_b128(const float* gsrc, float* ldst) {
    const unsigned lds_addr = (unsigned)(size_t)ldst;  // flat LDS addr low 32 bits = LDS byte offset
    asm volatile("global_load_async_to_lds_b128 %0, %1, off"
                 :
                 : "v"(lds_addr), "v"(gsrc)
                 : "memory");
}

__device__ __forceinline__ void wait_async0() {
#if __has_builtin(__builtin_amdgcn_s_wait_asynccnt)
    __builtin_amdgcn_s_wait_asynccnt(0);
#else
    asm volatile("s_wait_asynccnt 0x0" ::: "memory");
#endif
}

__global__ __launch_bounds__(128)
void topk_router_kernel(const float* __restrict__ x,
                        const float* __restrict__ W,
                        float* __restrict__ out_w,
                        long long* __restrict__ out_i)
{
    __shared__ float xs [2][TOK_PER_WG * LSTR];   // x tiles: 64 tokens x BK, double-buffered
    __shared__ float wsm[2][ROUTER_E   * LSTR];   // W tiles: 64 experts x BK, double-buffered

    const int t    = threadIdx.x;
    const int wave = t >> 5;          // 0..3
    const int lane = t & 31;
    const int tok0 = blockIdx.x * TOK_PER_WG;

    // A/B fragment addressing for 16x16x4 f32 WMMA:
    // lanes 0-15 hold K={0,1}, lanes 16-31 hold K={2,3}; row/col = lane&15.
    const int lrow = lane & 15;
    const int koff = (lane >> 4) << 1;

    // This thread's two (row, seg) chunks of each 64x64 tile.
    const int row0 = t >> 2;                 // 0..31
    const int row1 = (t + 128) >> 2;         // 32..63
    const int seg0 = (t & 3) << 2;
    const int seg1 = seg0;                   // (t+128)&3 == t&3

    v8f acc0 = {}, acc1 = {}, acc2 = {}, acc3 = {};

    // ---- prologue: issue stage 0 into buffer 0 ----
    {
        const float* xg = x + (size_t)tok0 * ROUTER_D;
        async_copy_b128(xg + (size_t)row0 * ROUTER_D + seg0, &xs[0][row0 * LSTR + seg0]);
        async_copy_b128(W  + (size_t)row0 * ROUTER_D + seg0, &wsm[0][row0 * LSTR + seg0]);
        async_copy_b128(xg + (size_t)row1 * ROUTER_D + seg1, &xs[0][row1 * LSTR + seg1]);
        async_copy_b128(W  + (size_t)row1 * ROUTER_D + seg1, &wsm[0][row1 * LSTR + seg1]);
    }

    for (int s = 0; s < NSTAGE; ++s) {
        const int p = s & 1;

        wait_async0();        // my async ops for stage s have landed in LDS
        __syncthreads();      // everyone's stage-s data visible; stage s-1 compute done

        if (s + 1 < NSTAGE) { // overlap stage s+1 HBM stream with stage s WMMA
            const int  kb = (s + 1) * BK;
            const int  q  = p ^ 1;
            const float* xg = x + (size_t)tok0 * ROUTER_D + kb;
            const float* wg = W + kb;
            async_copy_b128(xg + (size_t)row0 * ROUTER_D + seg0, &xs[q][row0 * LSTR + seg0]);
            async_copy_b128(wg + (size_t)row0 * ROUTER_D + seg0, &wsm[q][row0 * LSTR + seg0]);
            async_copy_b128(xg + (size_t)row1 * ROUTER_D + seg1, &xs[q][row1 * LSTR + seg1]);
            async_copy_b128(wg + (size_t)row1 * ROUTER_D + seg1, &wsm[q][row1 * LSTR + seg1]);
        }

        const float* xb = &xs[p][0];
        const float* wb = &wsm[p][0];

        // 16 K-steps of 4; 4 expert tiles each -> 64 v_wmma per stage.
        #pragma unroll
        for (int kk = 0; kk < BK; kk += 4) {
            const v2f a  = *(const v2f*)(&xb[(wave * 16 + lrow) * LSTR + kk + koff]);
            const v2f b0 = *(const v2f*)(&wb[( 0 + lrow) * LSTR + kk + koff]);
            const v2f b1 = *(const v2f*)(&wb[(16 + lrow) * LSTR + kk + koff]);
            const v2f b2 = *(const v2f*)(&wb[(32 + lrow) * LSTR + kk + koff]);
            const v2f b3 = *(const v2f*)(&wb[(48 + lrow) * LSTR + kk + koff]);
            acc0 = __builtin_amdgcn_wmma_f32_16x16x4_f32(false, a, false, b0, (short)0, acc0, false, false);
            acc1 = __builtin_amdgcn_wmma_f32_16x16x4_f32(false, a, false, b1, (short)0, acc1, false, false);
            acc2 = __builtin_amdgcn_wmma_f32_16x16x4_f32(false, a, false, b2, (short)0, acc2, false, false);
            acc3 = __builtin_amdgcn_wmma_f32_16x16x4_f32(false, a, false, b3, (short)0, acc3, false, false);
        }

        __syncthreads();      // all waves done with buffer p before it is re-filled
    }

    // Scatter C fragments (VGPR v -> token v or v+8; lane&15 -> expert col) to LDS.
    float* lg_base = &xs[0][0];
    #pragma unroll
    for (int v = 0; v < 8; ++v) {
        const int trow = wave * 16 + v + ((lane >> 4) << 3);
        lg_base[trow * LSTR +  0 + lrow] = acc0[v];
        lg_base[trow * LSTR + 16 + lrow] = acc1[v];
        lg_base[trow * LSTR + 32 + lrow] = acc2[v];
        lg_base[trow * LSTR + 48 + lrow] = acc3[v];
    }
    __syncthreads();

    // One thread per token: exact softmax over 64 experts, top-2, renormalize.
    if (t < TOK_PER_WG) {
        const float* lg = &lg_base[t * LSTR];

        float m = lg[0];
        #pragma unroll
        for (int e = 1; e < ROUTER_E; ++e) m = fmaxf(m, lg[e]);

        float z = 0.0f;
        float b1v = -3.402823466e+38f, b2v = -3.402823466e+38f;
        int   b1i = 0, b2i = 0;
        #pragma unroll
        for (int e = 0; e < ROUTER_E; ++e) {
            const float l = lg[e];
            z += __expf(l - m);
            if (l > b1v)      { b2v = b1v; b2i = b1i; b1v = l; b1i = e; }
            else if (l > b2v) { b2v = l;  b2i = e; }
        }
        const float p1 = __expf(b1v - m) / z;
        const float p2 = __expf(b2v - m) / z;
        const float s  = p1 + p2 + 1e-9f;

        const size_t tok = (size_t)tok0 + t;
        out_w[tok * 2 + 0] = p1 / s;
        out_w[tok * 2 + 1] = p2 / s;
        out_i[tok * 2 + 0] = (long long)b1i;
        out_i[tok * 2 + 1] = (long long)b2i;
    }
}

extern "C" void kernel_launch(void* const* d_in, const int* in_sizes, int n_in,
                              void* d_out, int out_size, void* d_ws, size_t ws_size,
                              hipStream_t stream)
{
    const float* x = (const float*)d_in[0];   // (B,T,D) f32
    const float* W = (const float*)d_in[1];   // (E,D)   f32

    const int N = in_sizes[0] / ROUTER_D;     // 16384 tokens

    float*     out_w = (float*)d_out;                       // (N,2) f32
    long long* out_i = (long long*)(out_w + (size_t)N * 2); // (N,2) i64

    dim3 grid(N / TOK_PER_WG);
    dim3 block(128);
    topk_router_kernel<<<grid, block, 0, stream>>>(x, W, out_w, out_i);
}